// EdgeScoring_72859825209919
// MI455X (gfx1250) — compile-verified
//
#include <hip/hip_runtime.h>
#include <hip/hip_bf16.h>

// Problem constants (match reference): B=16, N=128, C=1024, D=512
#define BB 16
#define NN 128
#define CC 1024
#define DD 512
#define MROWS (BB * NN)   // 2048 GEMM rows

typedef float v2f __attribute__((ext_vector_type(2)));
typedef float v8f __attribute__((ext_vector_type(8)));

#if __has_builtin(__builtin_amdgcn_tanhf)
  #define TANHF(x) __builtin_amdgcn_tanhf(x)
#elif __has_builtin(__builtin_amdgcn_tanh_f32)
  #define TANHF(x) __builtin_amdgcn_tanh_f32(x)
#else
  #define TANHF(x) tanhf(x)
#endif

#if __has_builtin(__builtin_amdgcn_tensor_load_to_lds) && \
    __has_builtin(__builtin_amdgcn_s_wait_tensorcnt)
  #define USE_TDM 1
#endif

// ---------------------------------------------------------------------------
// Kernel 1: projection GEMMs via V_WMMA_F32_16X16X4_F32.
//   Y[m, n] = sum_k X[m, k] * W[n, k]    (X: 2048x1024, W: 512x1024 row-major)
// One wave32 computes a 16x32 strip of Y (two 16x16 accumulators sharing the
// A fragment -> halves A load traffic, doubles WMMA density).
// blockIdx.z selects (Ua -> up) vs (Wa -> wc).
//
// ISA lane layouts for 16x16x4 f32 (CDNA5 ISA):
//   A 16x4:  lane L<16 holds A[M=L][K=k..k+1]; lane>=16 holds K=k+2..k+3
//   B 4x16:  lane L<16 holds B[K=k..k+1][N=L]; lane>=16 holds K=k+2..k+3
// Both map to ONE contiguous float2 (b64) load per lane per K-step, since K
// is the contiguous axis of both X and W.
//   D 16x16: VGPR r -> row (m0 + r + 8*laneHi), col (n0 + lane&15)
// ---------------------------------------------------------------------------
__global__ __launch_bounds__(128)
void EdgeScoring_proj_wmma(const float* __restrict__ X,
                           const float* __restrict__ Ua,
                           const float* __restrict__ Wa,
                           float* __restrict__ up,
                           float* __restrict__ wc)
{
    const float* __restrict__ W = blockIdx.z ? Wa : Ua;
    float*       __restrict__ Y = blockIdx.z ? wc : up;

    const int lane   = threadIdx.x & 31;
    const int wave   = threadIdx.x >> 5;           // 4 waves / block
    const int mt     = blockIdx.y * 4 + wave;      // 0..127  (2048/16)
    const int n0     = blockIdx.x * 32;            // 0..480  (two 16-col tiles)
    const int m0     = mt * 16;
    const int row    = lane & 15;
    const int laneHi = lane >> 4;                  // 0 or 1
    const int khalf  = laneHi * 2;                 // K sub-offset: 0 or 2

    const float* __restrict__ arow  = X + (size_t)(m0 + row) * CC + khalf;
    const float* __restrict__ brow0 = W + (size_t)(n0 + row) * CC + khalf;
    const float* __restrict__ brow1 = brow0 + (size_t)16 * CC;

    v8f acc0 = {};
    v8f acc1 = {};
    #pragma unroll 8
    for (int k = 0; k < CC; k += 4) {
        v2f a  = *(const v2f*)(arow + k);
        v2f b0 = *(const v2f*)(brow0 + k);
        v2f b1 = *(const v2f*)(brow1 + k);
        // 8 args: (neg_a, A, neg_b, B, c_mod, C, reuse_a, reuse_b)
        acc0 = __builtin_amdgcn_wmma_f32_16x16x4_f32(
            false, a, false, b0, (short)0, acc0, false, false);
        acc1 = __builtin_amdgcn_wmma_f32_16x16x4_f32(
            false, a, false, b1, (short)0, acc1, false, false);
    }

    // Scatter D tiles: VGPR r -> row m0 + r + 8*laneHi, col n0(+16) + (lane&15)
    float* ybase = Y + (size_t)(m0 + laneHi * 8) * DD + n0 + row;
    #pragma unroll
    for (int r = 0; r < 8; ++r) {
        ybase[(size_t)r * DD]      = acc0[r];
        ybase[(size_t)r * DD + 16] = acc1[r];
    }
}

// ---------------------------------------------------------------------------
// Kernel 2: fused score stage (never materializes h = (B,N,N,D)).
//   scores[b,i,j] = sum_d va[d] * tanh(up[b,i,d] + wc[b,j,d])
// One 256-thread block per (b, 16x16 ij tile). up/wc rows staged in LDS with a
// 4-float row pad (stride 516) so the 16 distinct wc rows land in distinct
// bank groups under float4 (b128) LDS reads.
//
// Staging uses the Tensor Data Mover when available: a D# describing a
// 16x512 f32 tile with hardware LDS padding (pad_interval=6 -> every
// 128 DWORDs, pad_amount=3 -> 4 DWORDs) reproduces the 516-float stride.
// ---------------------------------------------------------------------------
#define LSTRIDE 516  // 512 + 4-float pad, keeps 16B alignment of row starts

#ifdef USE_TDM
typedef unsigned int v4u __attribute__((ext_vector_type(4)));
typedef int          v8i __attribute__((ext_vector_type(8)));
typedef int          v4i __attribute__((ext_vector_type(4)));

// Issue one TDM load of a 16-row x 512-float tile (row stride 512 floats in
// global) into LDS at lds_dst, with 4-DWORD padding after every 128 DWORDs.
__device__ __forceinline__ void tdm_load_tile(const float* gsrc, void* lds_dst)
{
    const unsigned long long ga = (unsigned long long)(uintptr_t)gsrc;
    // Flat LDS pointer: high 32 bits = shared aperture, low 32 = LDS offset.
    const unsigned int lp = (unsigned int)(uintptr_t)lds_dst;

    v4u g0;
    g0.x = 1u;                                        // count=1, user-mode D#
    g0.y = lp;                                        // lds_addr (bytes)
    g0.z = (unsigned int)(ga & 0xffffffffu);          // global_addr[31:0]
    g0.w = (unsigned int)((ga >> 32) & 0x01ffffffu)   // global_addr[56:32]
         | 0x80000000u;                               // type = 2 ("image")

    v8i g1;
    g1[0] = (int)((2u << 16)      // data_size = 2 -> 4-byte elements
                | (1u << 20)      // pad_enable
                | (6u << 22)      // pad_interval: 2^(6+1) = 128 DWORDs
                | (3u << 25));    // pad_amount: 3+1 = 4 DWORDs
    g1[1] = (int)(512u << 16);    // tensor_dim0[15:0] = 512 (in [31:16])
    g1[2] = (int)(16u << 16);     // tensor_dim0 hi = 0; tensor_dim1 = 16
    g1[3] = (int)(512u << 16);    // tensor_dim1 hi = 0; tile_dim0 = 512
    g1[4] = 16;                   // tile_dim1 = 16; tile_dim2 = 0 (unused)
    g1[5] = 512;                  // tensor_dim0_stride[31:0] = 512 elements
    g1[6] = 0;                    // stride hi = 0; tensor_dim1_stride lo = 0
    g1[7] = 0;

    const v4i z4 = {0, 0, 0, 0};              // groups 2/3 unused (2D tile)
    const v8i z8 = {0, 0, 0, 0, 0, 0, 0, 0};  // trailing group (clang-23 form)
    // 6-arg amdgpu-toolchain signature: (v4u, v8i, v4i, v4i, v8i, i32 cpol)
    __builtin_amdgcn_tensor_load_to_lds(g0, g1, z4, z4, z8, 0);
}
#endif

__global__ __launch_bounds__(256)
void EdgeScoring_tanh_reduce(const float* __restrict__ up,
                             const float* __restrict__ wc,
                             const float* __restrict__ va,
                             float* __restrict__ out)
{
    __shared__ __align__(16) float sU[16][LSTRIDE];
    __shared__ __align__(16) float sW[16][LSTRIDE];
    __shared__ __align__(16) float sva[DD];

    const int b   = blockIdx.z;
    const int i0  = blockIdx.y * 16;
    const int j0  = blockIdx.x * 16;
    const int tid = threadIdx.x;

    const float* __restrict__ upb = up + ((size_t)b * NN + i0) * DD;
    const float* __restrict__ wcb = wc + ((size_t)b * NN + j0) * DD;

#ifdef USE_TDM
    // Wave 0 DMAs both tiles into LDS via the Tensor Data Mover; the barrier
    // releases the other waves once wave 0 has drained TENSORcnt.
    if (tid < 32) {
        tdm_load_tile(upb, (void*)&sU[0][0]);
        tdm_load_tile(wcb, (void*)&sW[0][0]);
        __builtin_amdgcn_s_wait_tensorcnt(0);
    }
#else
    // Cooperative stage: 16 rows x 512 floats each = 2048 float4 per tensor.
    for (int idx = tid; idx < 2048; idx += 256) {
        const int r  = idx >> 7;            // 128 float4 per row
        const int c4 = (idx & 127) * 4;
        float4 u = *(const float4*)(upb + (size_t)r * DD + c4);
        float4 w = *(const float4*)(wcb + (size_t)r * DD + c4);
        *(float4*)(&sU[r][c4]) = u;
        *(float4*)(&sW[r][c4]) = w;
    }
#endif
    if (tid < DD / 4)
        *(float4*)(&sva[tid * 4]) = *(const float4*)(va + tid * 4);
    __syncthreads();

    const int i = tid >> 4;   // 0..15
    const int j = tid & 15;   // 0..15

    float s = 0.0f;
    #pragma unroll 4
    for (int d = 0; d < DD; d += 4) {
        float4 u = *(const float4*)(&sU[i][d]);
        float4 w = *(const float4*)(&sW[j][d]);
        float4 v = *(const float4*)(&sva[d]);
        s += v.x * TANHF(u.x + w.x);
        s += v.y * TANHF(u.y + w.y);
        s += v.z * TANHF(u.z + w.z);
        s += v.w * TANHF(u.w + w.w);
    }

    out[(size_t)b * NN * NN + (size_t)(i0 + i) * NN + (j0 + j)] = s;
}

// ---------------------------------------------------------------------------
// Launch: kernel1 fills d_ws with up (4 MB) then wc (4 MB); kernel2 consumes.
// ---------------------------------------------------------------------------
extern "C" void kernel_launch(void* const* d_in, const int* in_sizes, int n_in,
                              void* d_out, int out_size, void* d_ws, size_t ws_size,
                              hipStream_t stream)
{
    const float* concepts = (const float*)d_in[0];   // (B,N,C)
    const float* Ua_w     = (const float*)d_in[1];   // (D,C)
    const float* Wa_w     = (const float*)d_in[2];   // (D,C)
    const float* va_w     = (const float*)d_in[3];   // (D,)
    float*       out      = (float*)d_out;           // (B,N,N)

    float* up = (float*)d_ws;                        // (B*N, D) = 4 MB
    float* wc = up + (size_t)MROWS * DD;             // (B*N, D) = 4 MB

    // GEMM grid: x = 32-col strips (16), y = M tile groups (32), z = {Ua, Wa}
    dim3 g1(DD / 32, MROWS / 16 / 4, 2);
    dim3 b1(128);
    EdgeScoring_proj_wmma<<<g1, b1, 0, stream>>>(concepts, Ua_w, Wa_w, up, wc);

    // Score grid: x = j tiles (8), y = i tiles (8), z = batch (16)
    dim3 g2(NN / 16, NN / 16, BB);
    dim3 b2(256);
    EdgeScoring_tanh_reduce<<<g2, b2, 0, stream>>>(up, wc, va_w, out);
}